// MultiTaskMazeQNetwork_73555609911661
// MI455X (gfx1250) — compile-verified
//
#include <hip/hip_runtime.h>
#include <stdint.h>

typedef float v2f  __attribute__((ext_vector_type(2)));
typedef float v8f  __attribute__((ext_vector_type(8)));
typedef unsigned int u32x4 __attribute__((ext_vector_type(4)));
typedef int  i32x4 __attribute__((ext_vector_type(4)));
typedef int  i32x8 __attribute__((ext_vector_type(8)));

#define NB    4096
#define FLATN 1024
#define NTASK 64

// ---------------------------------------------------------------------------
// TDM helpers (descriptor layout per CDNA5 ISA).
// LDS byte offsets assume the kernel's single __shared__ block starts at LDS 0.
// This toolchain's builtin takes 6 args: (g0, g1, g2, g3, g1b, cpol).
// ---------------------------------------------------------------------------
__device__ __forceinline__ void tdm_load_tile_2d(
    uint32_t lds_byte, const void* gptr,
    uint32_t tdim0, uint32_t tdim1, uint32_t tile0, uint32_t tile1,
    uint64_t stride0, int pad_en, uint32_t pad_int_code, uint32_t pad_amt_code)
{
  uint64_t ga = (uint64_t)(uintptr_t)gptr;
  u32x4 g0;
  g0[0] = 1u;                                             // count=1, normal mode
  g0[1] = lds_byte;                                       // lds_addr
  g0[2] = (uint32_t)ga;                                   // global_addr[31:0]
  g0[3] = ((uint32_t)(ga >> 32) & 0x1FFFFFFu) | (2u << 30); // addr[56:32] | type=2
  i32x8 g1;
  g1[0] = (int)((2u << 16) | (pad_en ? (1u << 20) : 0u) |
                (pad_int_code << 22) | (pad_amt_code << 25)); // data_size=4B
  g1[1] = (int)((tdim0 & 0xFFFFu) << 16);                 // tensor_dim0 lo
  g1[2] = (int)((tdim0 >> 16) | ((tdim1 & 0xFFFFu) << 16));
  g1[3] = (int)((tdim1 >> 16) | (tile0 << 16));           // tile_dim0
  g1[4] = (int)(tile1 & 0xFFFFu);                         // tile_dim1 (tile_dim2=0)
  g1[5] = (int)(uint32_t)stride0;                         // tensor_dim0_stride lo
  g1[6] = (int)((uint32_t)(stride0 >> 32) & 0xFFFFu);     // stride0 hi, stride1=0
  g1[7] = 0;
  i32x4 z4; z4[0] = z4[1] = z4[2] = z4[3] = 0;
  i32x8 z8; z8[0] = z8[1] = z8[2] = z8[3] = z8[4] = z8[5] = z8[6] = z8[7] = 0;
  __builtin_amdgcn_tensor_load_to_lds(g0, g1, z4, z4, z8, 0);
}

__device__ __forceinline__ void tdm_gather_rows(
    uint32_t lds_byte, const void* gptr, uint32_t row_len, uint32_t n_rows_tensor,
    int i0, int i1, int i2, int i3, int i4, int i5, int i6, int i7)
{
  uint64_t ga = (uint64_t)(uintptr_t)gptr;
  u32x4 g0;
  g0[0] = 1u | (1u << 30) | (1u << 31);                   // count=1, 32-bit idx, gather
  g0[1] = lds_byte;
  g0[2] = (uint32_t)ga;
  g0[3] = ((uint32_t)(ga >> 32) & 0x1FFFFFFu) | (2u << 30);
  i32x8 g1;
  g1[0] = (int)(2u << 16);                                // data_size=4B
  g1[1] = (int)((row_len & 0xFFFFu) << 16);               // tensor_dim0
  g1[2] = (int)((row_len >> 16) | ((n_rows_tensor & 0xFFFFu) << 16));
  g1[3] = (int)((n_rows_tensor >> 16) | ((row_len & 0xFFFFu) << 16)); // tile_dim0=row
  g1[4] = 8;                                              // tile_dim1 = #valid indices
  g1[5] = (int)row_len;                                   // tensor_dim0_stride
  g1[6] = 0;
  g1[7] = 0;
  i32x4 g2; g2[0] = i0; g2[1] = i1; g2[2] = i2; g2[3] = i3;
  i32x4 g3; g3[0] = i4; g3[1] = i5; g3[2] = i6; g3[3] = i7;
  i32x8 z8; z8[0] = z8[1] = z8[2] = z8[3] = z8[4] = z8[5] = z8[6] = z8[7] = 0;
  __builtin_amdgcn_tensor_load_to_lds(g0, g1, g2, g3, z8, 0);
}

// ---------------------------------------------------------------------------
// One 16x16 f32 output tile, K=1024, A in LDS (row stride 1024), B in global
// (row stride ldb). V_WMMA_F32_16X16X4_F32 per ISA VGPR layouts.
// ---------------------------------------------------------------------------
__device__ __forceinline__ v8f wmma_tile_k1024(const float* __restrict__ a_lds,
                                               const float* __restrict__ bcol,
                                               int ldb, int lane)
{
  v8f acc = {};
  const int m   = lane & 15;
  const int khi = (lane >> 4) << 1;        // lanes 0-15 -> K {0,1}; 16-31 -> {2,3}
  const float* ap = a_lds + m * FLATN + khi;
#pragma unroll 4
  for (int k0 = 0; k0 < FLATN; k0 += 4) {
    v2f a = *(const v2f*)(ap + k0);
    const float* bq = bcol + (size_t)(k0 + khi) * ldb;
    v2f b; b[0] = bq[0]; b[1] = bq[ldb];
    acc = __builtin_amdgcn_wmma_f32_16x16x4_f32(false, a, false, b,
                                                (short)0, acc, false, false);
  }
  return acc;
}

// ---------------------------------------------------------------------------
// K1: fused conv1(4x4,s1,p1)+relu -> conv2(4x4,s2,p2)+relu, one image / block.
// h1 (118 MB if materialized) lives only in LDS.
// ---------------------------------------------------------------------------
__global__ __launch_bounds__(256) void k_conv_fused(
    const float* __restrict__ x, const float* __restrict__ w1,
    const float* __restrict__ b1, const float* __restrict__ w2,
    const float* __restrict__ b2, float* __restrict__ flat)
{
  __shared__ float smem[1296 + 11552];     // x_s[18][18][4] , h1_s[19][19][32]
  float* x_s  = smem;
  float* h1_s = smem + 1296;
  const int b   = blockIdx.x;
  const int tid = threadIdx.x;

  for (int i = tid; i < 1296; i += 256) {  // zero pad border of x_s
    int row = i / 72, c = i % 72;
    if (row == 0 || row == 17 || c < 4 || c >= 68) x_s[i] = 0.0f;
  }
  for (int i = tid; i < 11552; i += 256) h1_s[i] = 0.0f;

  if (tid < 32) {
    // 16 rows x 64 floats; pad 8 DWORDs every 64 DWORDs -> LDS row stride 72.
    // Interior starts at x_s[1*72 + 1*4] = float index 76.
    tdm_load_tile_2d(76u * 4u, x + (size_t)b * 1024,
                     64, 16, 64, 16, 64, /*pad*/1, /*interval 64dw*/5, /*8dw*/7);
    __builtin_amdgcn_s_wait_tensorcnt(0);
  }
  __syncthreads();

  for (int o = tid; o < 15 * 15 * 32; o += 256) {   // conv1 -> h1 (padded by 2)
    int oc = o & 31, p = o >> 5;
    int ox = p % 15, oy = p / 15;
    float acc = b1[oc];
#pragma unroll
    for (int ky = 0; ky < 4; ++ky)
#pragma unroll
      for (int kx = 0; kx < 4; ++kx) {
        const float* xp = x_s + (oy + ky) * 72 + (ox + kx) * 4;
        const float* wp = w1 + ((ky * 4 + kx) * 4) * 32 + oc;
#pragma unroll
        for (int ic = 0; ic < 4; ++ic) acc += xp[ic] * wp[ic * 32];
      }
    h1_s[(oy + 2) * 608 + (ox + 2) * 32 + oc] = fmaxf(acc, 0.0f);
  }
  __syncthreads();

  for (int o = tid; o < 1024; o += 256) {           // conv2 -> flat[b]
    int oc = o & 15, p = o >> 4;
    int ox = p & 7, oy = p >> 3;
    float acc = b2[oc];
#pragma unroll
    for (int ky = 0; ky < 4; ++ky)
#pragma unroll
      for (int kx = 0; kx < 4; ++kx) {
        const float* hp = h1_s + (oy * 2 + ky) * 608 + (ox * 2 + kx) * 32;
        const float* wp = w2 + ((ky * 4 + kx) * 32) * 16 + oc;
        for (int ic = 0; ic < 32; ++ic) acc += hp[ic] * wp[ic * 16];
      }
    flat[(size_t)b * FLATN + o] = fmaxf(acc, 0.0f);
  }
}

// ---------------------------------------------------------------------------
// K2: shared = relu((flat @ se_w + se_b) @ sb_w + sb_b). 16 rows / block,
// 8 waves cover N=128 with WMMA f32 16x16x4 tiles; tiny 128x8 GEMM fused.
// ---------------------------------------------------------------------------
__global__ __launch_bounds__(256) void k_shared_mlp(
    const float* __restrict__ flat, const float* __restrict__ seW,
    const float* __restrict__ seB, const float* __restrict__ sbW,
    const float* __restrict__ sbB, float* __restrict__ shared8)
{
  __shared__ float smem[16 * FLATN];       // A tile; first 2048 reused as t[16][128]
  const int tid = threadIdx.x;
  const int m0  = blockIdx.x * 16;

  if (tid < 32) {
    tdm_load_tile_2d(0u, flat + (size_t)m0 * FLATN,
                     FLATN, 16, FLATN, 16, FLATN, 0, 0, 0);
    __builtin_amdgcn_s_wait_tensorcnt(0);
  }
  __syncthreads();

  const int wave = tid >> 5, lane = tid & 31;
  const int colBase = wave * 16;
  __builtin_prefetch(seW + colBase, 0, 1);
  v8f acc = wmma_tile_k1024(smem, seW + colBase + (lane & 15), 128, lane);
  __syncthreads();                         // done reading A before aliasing

  {
    int col = lane & 15, rbase = (lane >> 4) * 8;
#pragma unroll
    for (int v = 0; v < 8; ++v)
      smem[(rbase + v) * 128 + colBase + col] = acc[v] + seB[colBase + col];
  }
  __syncthreads();

  if (tid < 128) {                         // t[16][128] @ sb_w[128][8] + relu
    int m = tid >> 3, j = tid & 7;
    float s = sbB[j];
    const float* trow = smem + m * 128;
#pragma unroll 8
    for (int k = 0; k < 128; ++k) s += trow[k] * sbW[k * 8 + j];
    shared8[(size_t)(m0 + m) * 8 + j] = fmaxf(s, 0.0f);
  }
}

// ---------------------------------------------------------------------------
// K3: bucket examples by task (zero -> rank -> scan -> scatter).
// ---------------------------------------------------------------------------
__global__ void k_zero(int* cnt) { if (threadIdx.x < NTASK) cnt[threadIdx.x] = 0; }

__global__ __launch_bounds__(256) void k_rank(const int* __restrict__ tasks,
                                              int* cnt, int* rank) {
  int b = blockIdx.x * 256 + threadIdx.x;
  rank[b] = atomicAdd(&cnt[tasks[b]], 1);
}

__global__ void k_scan(const int* __restrict__ cnt, int* off) {
  if (threadIdx.x == 0) {
    int s = 0;
    for (int t = 0; t < NTASK; ++t) { off[t] = s; s += cnt[t]; }
  }
}

__global__ __launch_bounds__(256) void k_scatter(const int* __restrict__ tasks,
                                                 const int* __restrict__ off,
                                                 const int* __restrict__ rank,
                                                 int* bucket) {
  int b = blockIdx.x * 256 + threadIdx.x;
  bucket[off[tasks[b]] + rank[b]] = b;
}

// ---------------------------------------------------------------------------
// K4: per (task, 16-row chunk): TDM-gather 16 flat rows, WMMA 16x1024 @ 1024x32
// (tr_w[t]) + relu, then fused 40x4 head and scatter to d_out.
// ---------------------------------------------------------------------------
__global__ __launch_bounds__(64) void k_task_head(
    const float* __restrict__ flat, const float* __restrict__ shared8,
    const float* __restrict__ trW, const float* __restrict__ trB,
    const float* __restrict__ thW, const float* __restrict__ thB,
    const int* __restrict__ cnt, const int* __restrict__ off,
    const int* __restrict__ bucket, float* __restrict__ out)
{
  __shared__ float smem[16 * FLATN];       // gathered A; first 512 reused as rep
  const int t = blockIdx.y, chunk = blockIdx.x;
  const int n = cnt[t];
  if (chunk * 16 >= n) return;
  const int base  = off[t] + chunk * 16;
  const int nrows = min(16, n - chunk * 16);
  const int tid   = threadIdx.x;

  if (tid < 32) {
    int idx[16];
#pragma unroll
    for (int i = 0; i < 16; ++i) {
      int j = (i < nrows) ? i : 0;                         // clamp partial chunk
      idx[i] = __builtin_amdgcn_readfirstlane(bucket[base + j]);
    }
    tdm_gather_rows(0u, flat, FLATN, NB,
                    idx[0], idx[1], idx[2], idx[3], idx[4], idx[5], idx[6], idx[7]);
    tdm_gather_rows(8u * FLATN * 4u, flat, FLATN, NB,
                    idx[8], idx[9], idx[10], idx[11], idx[12], idx[13], idx[14], idx[15]);
    __builtin_amdgcn_s_wait_tensorcnt(0);
  }
  __syncthreads();

  const int wave = tid >> 5, lane = tid & 31;
  const int colBase = wave * 16;                           // 2 waves cover N=32
  v8f acc = wmma_tile_k1024(
      smem, trW + (size_t)t * FLATN * 32 + colBase + (lane & 15), 32, lane);
  __syncthreads();

  {
    int col = lane & 15, rbase = (lane >> 4) * 8;
#pragma unroll
    for (int v = 0; v < 8; ++v)
      smem[(rbase + v) * 32 + colBase + col] =
          fmaxf(acc[v] + trB[t * 32 + colBase + col], 0.0f);
  }
  __syncthreads();

  {
    int r = tid >> 2, a = tid & 3;
    if (r < nrows) {
      int bi = bucket[base + r];
      float q = thB[t * 4 + a];
      const float* w  = thW + (size_t)t * 40 * 4 + a;
      const float* sh = shared8 + (size_t)bi * 8;
#pragma unroll
      for (int j = 0; j < 8; ++j)  q += sh[j] * w[j * 4];
      const float* rep = smem + r * 32;
#pragma unroll
      for (int h = 0; h < 32; ++h) q += rep[h] * w[(8 + h) * 4];
      out[(size_t)bi * 4 + a] = q;
    }
  }
}

// ---------------------------------------------------------------------------
extern "C" void kernel_launch(void* const* d_in, const int* in_sizes, int n_in,
                              void* d_out, int out_size, void* d_ws, size_t ws_size,
                              hipStream_t stream) {
  const float* x      = (const float*)d_in[0];
  const int*   tasks  = (const int*)  d_in[1];
  const float* w1     = (const float*)d_in[2];
  const float* b1     = (const float*)d_in[3];
  const float* w2     = (const float*)d_in[4];
  const float* b2     = (const float*)d_in[5];
  const float* seW    = (const float*)d_in[6];
  const float* seB    = (const float*)d_in[7];
  const float* sbW    = (const float*)d_in[8];
  const float* sbB    = (const float*)d_in[9];
  const float* trW    = (const float*)d_in[10];
  const float* trB    = (const float*)d_in[11];
  const float* thW    = (const float*)d_in[12];
  const float* thB    = (const float*)d_in[13];

  float* flat    = (float*)d_ws;
  float* shared8 = flat + (size_t)NB * FLATN;
  int*   cnt     = (int*)(shared8 + (size_t)NB * 8);
  int*   off     = cnt + NTASK;
  int*   rank    = off + NTASK;
  int*   bucket  = rank + NB;

  k_zero<<<1, 64, 0, stream>>>(cnt);
  k_conv_fused<<<NB, 256, 0, stream>>>(x, w1, b1, w2, b2, flat);
  k_shared_mlp<<<NB / 16, 256, 0, stream>>>(flat, seW, seB, sbW, sbB, shared8);
  k_rank<<<NB / 256, 256, 0, stream>>>(tasks, cnt, rank);
  k_scan<<<1, 64, 0, stream>>>(cnt, off);
  k_scatter<<<NB / 256, 256, 0, stream>>>(tasks, off, rank, bucket);
  k_task_head<<<dim3(NB / 16, NTASK), 64, 0, stream>>>(
      flat, shared8, trW, trB, thW, thB, cnt, off, bucket, (float*)d_out);
}